// GTEProgramClassification_27986006900835
// MI455X (gfx1250) — compile-verified
//
#include <hip/hip_runtime.h>

typedef __attribute__((ext_vector_type(2))) float v2f;
typedef __attribute__((ext_vector_type(4))) float v4f;
typedef __attribute__((ext_vector_type(8))) float v8f;

constexpr int D     = 256;   // hidden dim
constexpr int C_OUT = 104;   // classes
constexpr int LDA   = 260;   // LDS row stride (260 % 64 == 4 -> conflict-free frag reads)

// ---------------------------------------------------------------------------
// Kernel 1: segment boundaries on sorted dst_idx. Every dst has >=1 edge.
// ---------------------------------------------------------------------------
__global__ void seg_bounds_kernel(const int* __restrict__ dst,
                                  int* __restrict__ seg_start,
                                  int E, int n_dst) {
    int e = blockIdx.x * blockDim.x + threadIdx.x;
    if (e >= E) return;
    int d = dst[e];
    if (e == 0 || dst[e - 1] != d) seg_start[d] = e;
    if (e == E - 1) seg_start[n_dst] = E;
}

// ---------------------------------------------------------------------------
// Kernel 2: one wave32 per dst node. lane owns 8 contiguous columns.
// child_sum = sum of msgs excluding last; last_msg = final msg; deg.
// msgs[e] = emb[token_id[src_idx[e]]]
// ---------------------------------------------------------------------------
__global__ __launch_bounds__(256)
void aggregate_kernel(const float* __restrict__ emb,
                      const int* __restrict__ token_id,
                      const int* __restrict__ src_idx,
                      const int* __restrict__ seg_start,
                      float* __restrict__ child_sum,
                      float* __restrict__ last_msg,
                      int* __restrict__ deg_out,
                      int n_dst) {
    const int wave = threadIdx.x >> 5;
    const int lane = threadIdx.x & 31;
    const int d = blockIdx.x * (blockDim.x >> 5) + wave;
    if (d >= n_dst) return;

    const int s    = seg_start[d];
    const int epos = seg_start[d + 1];
    const int c    = lane * 8;

    v4f acc0 = {0.f, 0.f, 0.f, 0.f};
    v4f acc1 = {0.f, 0.f, 0.f, 0.f};
    for (int e = s; e < epos - 1; ++e) {
        const int row = token_id[src_idx[e]];
        const v4f* p = (const v4f*)(emb + (size_t)row * D + c);
        acc0 += p[0];
        acc1 += p[1];
    }
    const int lrow = token_id[src_idx[epos - 1]];
    const v4f* lp = (const v4f*)(emb + (size_t)lrow * D + c);
    v4f l0 = lp[0], l1 = lp[1];

    *(v4f*)(child_sum + (size_t)d * D + c)     = acc0;
    *(v4f*)(child_sum + (size_t)d * D + c + 4) = acc1;
    *(v4f*)(last_msg  + (size_t)d * D + c)     = l0;
    *(v4f*)(last_msg  + (size_t)d * D + c + 4) = l1;
    if (lane == 0) deg_out[d] = epos - s;
}

// ---------------------------------------------------------------------------
// Kernel 3: fused RNN-cell GEMM + classifier GEMM on the WMMA pipe.
// 512 threads = 16 waves per block; block owns a 16-row strip of dst nodes.
// Phase 1: wave w computes H[:, 16w:16w+16] = child_sum_strip @ W^T via
//          64x v_wmma_f32_16x16x4_f32, then ft = gate(last, relu(H+b), deg)
//          into LDS.
// Phase 2: waves 0..6 compute out_strip = ft_strip @ Wc^T + bc (C=104).
// ---------------------------------------------------------------------------
__global__ __launch_bounds__(512)
void fused_gemm_kernel(const float* __restrict__ child_sum,
                       const float* __restrict__ last_msg,
                       const int* __restrict__ deg,
                       const float* __restrict__ W,
                       const float* __restrict__ b,
                       const float* __restrict__ Wc,
                       const float* __restrict__ bc,
                       float* __restrict__ out,
                       int n_dst) {
    __shared__ float sA[16 * LDA];   // child_sum strip
    __shared__ float sL[16 * LDA];   // last_msg strip
    __shared__ float sF[16 * LDA];   // ft strip (phase-1 result)
    __shared__ int   sDeg[16];

    const int t    = threadIdx.x;
    const int wave = t >> 5;
    const int lane = t & 31;
    const int m0   = blockIdx.x * 16;

    // cooperative stage of the 16xD activation strips
    for (int i = t; i < 16 * D; i += 512) {
        const int r  = i >> 8;          // D == 256
        const int cc = i & (D - 1);
        const int gm = (m0 + r < n_dst) ? (m0 + r) : (n_dst - 1);
        sA[r * LDA + cc] = child_sum[(size_t)gm * D + cc];
        sL[r * LDA + cc] = last_msg[(size_t)gm * D + cc];
    }
    if (t < 16) {
        const int gm = (m0 + t < n_dst) ? (m0 + t) : (n_dst - 1);
        sDeg[t] = deg[gm];
    }
    __syncthreads();

    // WMMA fragment indexing (ISA 7.12.2, f32 16x16x4):
    //   A: lanes 0-15 -> M=lane, K={k,k+1}; lanes 16-31 -> M=lane-16, K={k+2,k+3}
    //   B: lanes 0-15 -> N=lane, K={k,k+1}; lanes 16-31 -> N=lane-16, K={k+2,k+3}
    //   D: vgpr r -> M = r + 8*(lane>>4), N = lane&15
    const int col  = lane & 15;
    const int koff = (lane >> 4) << 1;   // 0 or 2

    // ---- Phase 1: H = child_sum @ W^T ; ft = gate(last, relu(H + b)) ----
    {
        const int n = wave * 16 + col;               // output column (0..255)
        const float* wrow = W + (size_t)n * D;       // B[k][n] = W[n][k]
        v8f acc = {};
        for (int k = 0; k < D; k += 4) {
            v2f a, bb;
            a.x  = sA[col * LDA + k + koff];
            a.y  = sA[col * LDA + k + koff + 1];
            bb.x = wrow[k + koff];
            bb.y = wrow[k + koff + 1];
            acc = __builtin_amdgcn_wmma_f32_16x16x4_f32(
                      false, a, false, bb, (short)0, acc, false, false);
        }
        const float bn = b[n];
        const int half = lane >> 4;
        #pragma unroll
        for (int r = 0; r < 8; ++r) {
            const int lr = r + half * 8;             // local row in strip
            const float h     = fmaxf(acc[r] + bn, 0.0f);
            const float lastv = sL[lr * LDA + n];
            sF[lr * LDA + n]  = (sDeg[lr] == 1) ? lastv : (lastv + h);
        }
    }
    __syncthreads();

    // ---- Phase 2: out = ft @ Wc^T + bc  (C=104 -> 7 column tiles) ----
    if (wave < 7) {                                   // wave-uniform branch
        const int n = wave * 16 + col;
        const bool valid = (n < C_OUT);
        const float sel = valid ? 1.0f : 0.0f;
        const float* wcrow = Wc + (size_t)(valid ? n : 0) * D;  // B[k][n]=Wc[n][k]
        v8f acc = {};
        for (int k = 0; k < D; k += 4) {
            v2f a, bb;
            a.x  = sF[col * LDA + k + koff];
            a.y  = sF[col * LDA + k + koff + 1];
            bb.x = wcrow[k + koff] * sel;             // branch-free mask: full EXEC at WMMA
            bb.y = wcrow[k + koff + 1] * sel;
            acc = __builtin_amdgcn_wmma_f32_16x16x4_f32(
                      false, a, false, bb, (short)0, acc, false, false);
        }
        if (valid) {
            const float bcn = bc[n];
            const int half = lane >> 4;
            #pragma unroll
            for (int r = 0; r < 8; ++r) {
                const int lr = r + half * 8;
                if (m0 + lr < n_dst)
                    out[(size_t)(m0 + lr) * C_OUT + n] = acc[r] + bcn;
            }
        }
    }
}

// ---------------------------------------------------------------------------
// Host-side launcher
// inputs (setup_inputs order): emb, W, b, Wc, bc, token_id, src_idx, dst_idx
// ---------------------------------------------------------------------------
extern "C" void kernel_launch(void* const* d_in, const int* in_sizes, int n_in,
                              void* d_out, int out_size, void* d_ws, size_t ws_size,
                              hipStream_t stream) {
    const float* emb      = (const float*)d_in[0];
    const float* W        = (const float*)d_in[1];
    const float* b        = (const float*)d_in[2];
    const float* Wc       = (const float*)d_in[3];
    const float* bc       = (const float*)d_in[4];
    const int*   token_id = (const int*)d_in[5];
    const int*   src_idx  = (const int*)d_in[6];
    const int*   dst_idx  = (const int*)d_in[7];

    const int E     = in_sizes[6];
    const int n_dst = out_size / C_OUT;

    // workspace carve-up (256B aligned)
    char* ws = (char*)d_ws;
    size_t off = 0;
    int* seg_start = (int*)(ws + off);
    off += (((size_t)(n_dst + 1) * sizeof(int)) + 255) & ~(size_t)255;
    int* deg = (int*)(ws + off);
    off += (((size_t)n_dst * sizeof(int)) + 255) & ~(size_t)255;
    float* child_sum = (float*)(ws + off);
    off += (size_t)n_dst * D * sizeof(float);
    float* last_msg = (float*)(ws + off);

    // 1) segment boundaries
    seg_bounds_kernel<<<(E + 255) / 256, 256, 0, stream>>>(dst_idx, seg_start, E, n_dst);

    // 2) segment aggregation: one wave32 per dst, 8 waves per block
    aggregate_kernel<<<(n_dst + 7) / 8, 256, 0, stream>>>(
        emb, token_id, src_idx, seg_start, child_sum, last_msg, deg, n_dst);

    // 3) fused WMMA GEMMs: 16 dst rows per block
    fused_gemm_kernel<<<(n_dst + 15) / 16, 512, 0, stream>>>(
        child_sum, last_msg, deg, W, b, Wc, bc, (float*)d_out, n_dst);
}